// RNN_48954037240221
// MI455X (gfx1250) — compile-verified
//
#include <hip/hip_runtime.h>

// Problem sizes (fixed by the reference).
#define B_  64
#define T_  512
#define D_  1024
#define H_  1024

typedef __attribute__((ext_vector_type(2))) float v2f;
typedef __attribute__((ext_vector_type(8))) float v8f;

__device__ __forceinline__ v8f wmma_f32(v2f a, v2f b, v8f c) {
  // D = A(16x4 f32) * B(4x16 f32) + C(16x16 f32)
  return __builtin_amdgcn_wmma_f32_16x16x4_f32(
      /*neg_a=*/false, a, /*neg_b=*/false, b,
      /*c_mod=*/(short)0, c, /*reuse_a=*/false, /*reuse_b=*/false);
}

__device__ __forceinline__ float fast_tanh(float x) {
  x = fminf(fmaxf(x, -15.0f), 15.0f);
  float e = __expf(2.0f * x);
  return (e - 1.0f) / (e + 1.0f);
}

// ---------------------------------------------------------------------------
// Kernel 0: zero the per-row-group barrier counters (d_ws is poisoned 0xAA).
// ---------------------------------------------------------------------------
__global__ void rnn_init_kernel(int* cnt) {
  if (threadIdx.x < 64) cnt[threadIdx.x] = 0;
}

// ---------------------------------------------------------------------------
// Kernel 1: xp = x @ Wx + b  -> d_out (consumed in place by the scan).
// M = B*T = 32768, N = H = 1024, K = D = 1024.
// 256 threads (8 waves), 128x128 tile, K-chunks of 16, f32 WMMA 16x16x4.
// Double-buffered LDS; both A ([row][k]) and B^T ([col][k]) use stride 20
// so every fragment is one conflict-free ds_load_b64.
// ---------------------------------------------------------------------------
#define TM 128
#define TN 128
#define TK 16
#define LDT 20   // tile k-stride in floats (16 + pad; rows 16B-aligned)

__global__ __launch_bounds__(256)
void rnn_xproj_kernel(const float* __restrict__ x, const float* __restrict__ Wx,
                      const float* __restrict__ bias, float* __restrict__ out) {
  __shared__ float sA[2][TM * LDT];    // A tile:  [row][k]
  __shared__ float sBt[2][TN * LDT];   // B tile transposed: [col][k]

  const int tid  = threadIdx.x;
  const int lane = tid & 31;
  const int wave = tid >> 5;      // 0..7
  const int mGrp = wave >> 2;     // 0..1 : 4 M-tiles each
  const int nGrp = wave & 3;      // 0..3 : 2 N-tiles each
  const int mBase = blockIdx.y * TM;
  const int nBase = blockIdx.x * TN;
  const int laneHalf = lane >> 4; // 0/1 -> K pair (0,1) or (2,3)
  const int lr = lane & 15;

  // Per-thread staging coordinates.
  const int aRow0 = (tid + 0)   >> 2, aC0 = ((tid + 0)   & 3) << 2;
  const int aRow1 = (tid + 256) >> 2, aC1 = ((tid + 256) & 3) << 2;
  const int bK0   = (tid + 0)   >> 5, bC0 = ((tid + 0)   & 31) << 2;
  const int bK1   = (tid + 256) >> 5, bC1 = ((tid + 256) & 31) << 2;

  v8f acc[4][2];
#pragma unroll
  for (int i = 0; i < 4; ++i)
#pragma unroll
    for (int j = 0; j < 2; ++j) acc[i][j] = {};

  float4 ra0, ra1, rb0, rb1;

  // Prologue: stage chunk 0.
  ra0 = *(const float4*)(x + (size_t)(mBase + aRow0) * D_ + 0 + aC0);
  ra1 = *(const float4*)(x + (size_t)(mBase + aRow1) * D_ + 0 + aC1);
  rb0 = *(const float4*)(Wx + (size_t)(0 + bK0) * H_ + nBase + bC0);
  rb1 = *(const float4*)(Wx + (size_t)(0 + bK1) * H_ + nBase + bC1);
  *(float4*)(&sA[0][aRow0 * LDT + aC0]) = ra0;
  *(float4*)(&sA[0][aRow1 * LDT + aC1]) = ra1;
  sBt[0][(bC0 + 0) * LDT + bK0] = rb0.x;
  sBt[0][(bC0 + 1) * LDT + bK0] = rb0.y;
  sBt[0][(bC0 + 2) * LDT + bK0] = rb0.z;
  sBt[0][(bC0 + 3) * LDT + bK0] = rb0.w;
  sBt[0][(bC1 + 0) * LDT + bK1] = rb1.x;
  sBt[0][(bC1 + 1) * LDT + bK1] = rb1.y;
  sBt[0][(bC1 + 2) * LDT + bK1] = rb1.z;
  sBt[0][(bC1 + 3) * LDT + bK1] = rb1.w;
  __syncthreads();

  int cur = 0;
  for (int kc = 0; kc < D_; kc += TK) {
    const bool more = (kc + TK) < D_;
    // Issue global loads for the next chunk (overlap with WMMA below).
    if (more) {
      const int kn = kc + TK;
      ra0 = *(const float4*)(x + (size_t)(mBase + aRow0) * D_ + kn + aC0);
      ra1 = *(const float4*)(x + (size_t)(mBase + aRow1) * D_ + kn + aC1);
      rb0 = *(const float4*)(Wx + (size_t)(kn + bK0) * H_ + nBase + bC0);
      rb1 = *(const float4*)(Wx + (size_t)(kn + bK1) * H_ + nBase + bC1);
    }
    // Compute current chunk: 4 K-steps x (4 M-tiles x 2 N-tiles) = 32 WMMA.
    const float* __restrict__ cA = sA[cur];
    const float* __restrict__ cB = sBt[cur];
#pragma unroll
    for (int kk = 0; kk < TK; kk += 4) {
      const int kOff = kk + (laneHalf << 1);
      v2f afr[4], bfr[2];
#pragma unroll
      for (int i = 0; i < 4; ++i)
        afr[i] = *(const v2f*)(cA + ((mGrp * 4 + i) * 16 + lr) * LDT + kOff);
#pragma unroll
      for (int j = 0; j < 2; ++j)
        bfr[j] = *(const v2f*)(cB + ((nGrp * 2 + j) * 16 + lr) * LDT + kOff);
#pragma unroll
      for (int i = 0; i < 4; ++i)
#pragma unroll
        for (int j = 0; j < 2; ++j)
          acc[i][j] = wmma_f32(afr[i], bfr[j], acc[i][j]);
    }
    // Stage next chunk into the other buffer.
    if (more) {
      const int nxt = cur ^ 1;
      *(float4*)(&sA[nxt][aRow0 * LDT + aC0]) = ra0;
      *(float4*)(&sA[nxt][aRow1 * LDT + aC1]) = ra1;
      sBt[nxt][(bC0 + 0) * LDT + bK0] = rb0.x;
      sBt[nxt][(bC0 + 1) * LDT + bK0] = rb0.y;
      sBt[nxt][(bC0 + 2) * LDT + bK0] = rb0.z;
      sBt[nxt][(bC0 + 3) * LDT + bK0] = rb0.w;
      sBt[nxt][(bC1 + 0) * LDT + bK1] = rb1.x;
      sBt[nxt][(bC1 + 1) * LDT + bK1] = rb1.y;
      sBt[nxt][(bC1 + 2) * LDT + bK1] = rb1.z;
      sBt[nxt][(bC1 + 3) * LDT + bK1] = rb1.w;
    }
    __syncthreads();
    cur ^= 1;
  }

  // Store with bias. C/D layout: VGPR r -> (M = r + 8*laneHalf, N = lr).
#pragma unroll
  for (int i = 0; i < 4; ++i) {
#pragma unroll
    for (int j = 0; j < 2; ++j) {
      const int col = nBase + (nGrp * 2 + j) * 16 + lr;
      const float bv = bias[col];
#pragma unroll
      for (int r = 0; r < 8; ++r) {
        const int mrow = mBase + (mGrp * 4 + i) * 16 + r + (laneHalf << 3);
        out[(size_t)mrow * H_ + col] = acc[i][j][r] + bv;
      }
    }
  }
}

// ---------------------------------------------------------------------------
// Kernel 2: recurrent scan, in place on d_out.
// 64 WGs = 4 row-groups (16 batch rows = one WMMA M-tile) x 16 col-WGs
// (64 H-columns each). Wh column slice transposed in LDS (64 x 1026 f32 =
// 256.5 KB, using CDNA5's 320 KB LDS/WGP). Per step each wave accumulates a
// 16x16 tile over K=1024 with FOUR independent accumulators to break the
// WMMA dependency chain, then h_t = tanh(xp_t + acc) overwrites xp in d_out.
// ---------------------------------------------------------------------------
#define SLD 1026  // LDS k-stride (floats) for transposed Wh slice

__global__ __launch_bounds__(128)
void rnn_scan_kernel(const float* __restrict__ Wh, float* __restrict__ out,
                     int* __restrict__ barrier_cnt) {
  extern __shared__ float sWhT[];  // [64 cols][SLD] : sWhT[c][k] = Wh[k][colBase+c]

  const int tid  = threadIdx.x;
  const int lane = tid & 31;
  const int wave = tid >> 5;           // 0..3  (one 16x16 col tile per wave)
  const int grp  = blockIdx.x >> 4;    // 0..3  row group (batch rows 16*grp..)
  const int cw   = blockIdx.x & 15;    // 0..15 column workgroup
  const int colBase = cw * 64;
  const int laneHalf = lane >> 4;
  const int lr = lane & 15;

  // Stage Wh[:, colBase:colBase+64] transposed into LDS (one-time, L2-fed).
  for (int i = tid; i < (H_ * 64) / 4; i += 128) {
    int k  = i >> 4;
    int c4 = (i & 15) << 2;
    float4 v = *(const float4*)(Wh + (size_t)k * H_ + colBase + c4);
    sWhT[(c4 + 0) * SLD + k] = v.x;
    sWhT[(c4 + 1) * SLD + k] = v.y;
    sWhT[(c4 + 2) * SLD + k] = v.z;
    sWhT[(c4 + 3) * SLD + k] = v.w;
  }
  __syncthreads();

  const int col = colBase + wave * 16 + lr;         // this lane's N column
  const float* whCol = sWhT + (wave * 16 + lr) * SLD + (laneHalf << 1);
  int* cnt = barrier_cnt + grp;

  for (int t = 0; t < T_; ++t) {
    // Prefetch the xp lines we will read+overwrite after the GEMM
    // (global_prefetch_b8; overlaps with the 256-WMMA K loop).
#pragma unroll
    for (int r = 0; r < 8; ++r) {
      const int b = grp * 16 + r + (laneHalf << 3);
      __builtin_prefetch(out + ((size_t)b * T_ + t) * H_ + col, 0, 3);
    }

    v8f a0 = {}, a1 = {}, a2 = {}, a3 = {};
    if (t > 0) {
      // A-matrix rows: batch b = grp*16 + lr, K runs along h_{t-1}.
      const float* hprev =
          out + ((size_t)(grp * 16 + lr) * T_ + (t - 1)) * H_ + (laneHalf << 1);
#pragma unroll 4
      for (int kk = 0; kk < H_; kk += 16) {
        v2f ha0 = *(const v2f*)(hprev + kk + 0);
        v2f ha1 = *(const v2f*)(hprev + kk + 4);
        v2f ha2 = *(const v2f*)(hprev + kk + 8);
        v2f ha3 = *(const v2f*)(hprev + kk + 12);
        v2f wb0 = *(const v2f*)(whCol + kk + 0);
        v2f wb1 = *(const v2f*)(whCol + kk + 4);
        v2f wb2 = *(const v2f*)(whCol + kk + 8);
        v2f wb3 = *(const v2f*)(whCol + kk + 12);
        a0 = wmma_f32(ha0, wb0, a0);
        a1 = wmma_f32(ha1, wb1, a1);
        a2 = wmma_f32(ha2, wb2, a2);
        a3 = wmma_f32(ha3, wb3, a3);
      }
    }
    const v8f acc = (a0 + a1) + (a2 + a3);

    // h_t = tanh(xp_t + acc), overwriting xp in place in d_out.
#pragma unroll
    for (int r = 0; r < 8; ++r) {
      const int b = grp * 16 + r + (laneHalf << 3);
      float* p = out + ((size_t)b * T_ + t) * H_ + col;
      *p = fast_tanh(acc[r] + *p);
    }

    // Row-group barrier: 16 col-WGs arrive, then proceed to step t+1.
    __threadfence();
    __syncthreads();
    if (tid == 0) {
      __hip_atomic_fetch_add(cnt, 1, __ATOMIC_RELEASE, __HIP_MEMORY_SCOPE_AGENT);
      const int target = 16 * (t + 1);
      while (__hip_atomic_load(cnt, __ATOMIC_ACQUIRE, __HIP_MEMORY_SCOPE_AGENT) < target)
        __builtin_amdgcn_s_sleep(1);
    }
    __syncthreads();
  }
}

// ---------------------------------------------------------------------------
extern "C" void kernel_launch(void* const* d_in, const int* in_sizes, int n_in,
                              void* d_out, int out_size, void* d_ws, size_t ws_size,
                              hipStream_t stream) {
  const float* x    = (const float*)d_in[0];  // [B,T,D]
  const float* Wx   = (const float*)d_in[1];  // [D,H]
  const float* Wh   = (const float*)d_in[2];  // [H,H]
  const float* bias = (const float*)d_in[3];  // [H]
  float* out = (float*)d_out;                 // [B,T,H]
  int* cnt = (int*)d_ws;

  rnn_init_kernel<<<1, 64, 0, stream>>>(cnt);

  dim3 g1(H_ / TN, (B_ * T_) / TM);           // (8, 256) tiles
  rnn_xproj_kernel<<<g1, 256, 0, stream>>>(x, Wx, bias, out);

  size_t smem = (size_t)64 * SLD * sizeof(float);  // 256.5 KB of CDNA5 LDS
  rnn_scan_kernel<<<64, 128, smem, stream>>>(Wh, out, cnt);
}